// OnsetsMAE_57604101373969
// MI455X (gfx1250) — compile-verified
//
#include <hip/hip_runtime.h>
#include <hip/hip_bf16.h>
#include <stdint.h>

#define ROWS   32
#define LLEN   131072
#define PADW   12            // WIDTH=25, PAD=12
#define CHUNK  8192
#define TPB    1024
#define MAXP   10752         // >= floor((L-1)/13)+1 = 10083 max peaks per row
#define NEGINF (-3.4028234663852886e38f)

#if __has_builtin(__builtin_amdgcn_s_wait_asynccnt)
#define WAIT_ASYNC(n) __builtin_amdgcn_s_wait_asynccnt(n)
#else
#define WAIT_ASYNC(n) asm volatile("s_wait_asynccnt %0" :: "i"(n) : "memory")
#endif

// ---------------------------------------------------------------------------
// Issue one chunk's staging: 2 async b128 DMA ops per thread (interior) plus
// a 24-element halo via the regular path, clamped to -FLT_MAX at row edges.
// ---------------------------------------------------------------------------
__device__ __forceinline__ void issue_chunk(const float* __restrict__ src,
                                            int cs, float* tilebuf, int tid) {
#pragma unroll
  for (int c = 0; c < CHUNK / (TPB * 4); ++c) {
    const int idx = (c * TPB + tid) * 4;
    // low 32 bits of a generic LDS pointer == workgroup-relative LDS byte addr
    unsigned ldsOff = (unsigned)(uintptr_t)(&tilebuf[PADW + idx]);
    unsigned long long ga = (unsigned long long)(uintptr_t)(src + cs + idx);
    asm volatile("global_load_async_to_lds_b128 %0, %1, off"
                 :: "v"(ldsOff), "v"(ga) : "memory");
  }
  if (tid < 2 * PADW) {
    const int  li   = (tid < PADW) ? tid : (PADW + CHUNK + (tid - PADW));
    const long gidx = (long)cs - PADW +
                      ((tid < PADW) ? (long)tid : (long)(CHUNK + (tid - PADW)));
    tilebuf[li] = (gidx >= 0 && gidx < LLEN) ? src[gidx] : NEGINF;
  }
}

// ---------------------------------------------------------------------------
// Kernel A: streaming peak detection + ordered compaction.
// grid = (32 rows, 2 signals), block = 1024 (32 waves, wave32).
// Double-buffered async global->LDS staging pipelined on ASYNCcnt: chunk k+1
// is in flight (2 outstanding async ops/wave) while chunk k is computed.
// ---------------------------------------------------------------------------
__global__ __launch_bounds__(TPB) void onsets_peaks_kernel(
    const float* __restrict__ preds, const float* __restrict__ labels,
    float* __restrict__ posOut, int* __restrict__ counts) {
  const int row = blockIdx.x;
  const int sig = blockIdx.y;
  const float* src = (sig == 0 ? preds : labels) + (size_t)row * LLEN;
  float* outPos = posOut + ((size_t)sig * ROWS + row) * MAXP;

  __shared__ float tile[2][CHUNK + 2 * PADW];
  __shared__ int waveCnt[32];

  const int tid  = threadIdx.x;
  const int lane = tid & 31;
  const int wid  = tid >> 5;

  int running = 0;  // identical in every thread

  issue_chunk(src, 0, tile[0], tid);  // prologue: chunk 0 into buffer 0

  const int NCH = LLEN / CHUNK;
#pragma unroll 1
  for (int ci = 0; ci < NCH; ++ci) {
    const int cs = ci * CHUNK;
    float* cur = tile[ci & 1];

    if (ci + 1 < NCH) {
      issue_chunk(src, cs + CHUNK, tile[(ci + 1) & 1], tid);
      WAIT_ASYNC(2);  // retire the older chunk's 2 ops; keep next 2 in flight
    } else {
      WAIT_ASYNC(0);
    }
    __syncthreads();  // cur fully staged (async + halo ds stores) for all waves

    // --- 8 sub-rounds, one element per thread, globally ascending order ----
#pragma unroll 1
    for (int e = 0; e < CHUNK / TPB; ++e) {
      const int lt = e * TPB + tid;
      const float v = cur[PADW + lt];
      float mL = cur[lt];                 // t-12
      float mR = cur[PADW + lt + 1];      // t+1
#pragma unroll
      for (int d = 1; d < PADW; ++d) {
        mL = fmaxf(mL, cur[lt + d]);               // t-12 .. t-1
        mR = fmaxf(mR, cur[PADW + lt + 1 + d]);    // t+1  .. t+12
      }
      const bool peak = (mL < v) && (mR <= v);      // first-argmax semantics

      unsigned long long m = __ballot(peak);        // wave32: low 32 bits
      const int pre = __popcll(m & ((1ull << lane) - 1ull));
      if (lane == 0) waveCnt[wid] = __popcll(m);
      __syncthreads();
      int base = running, tot = 0;
#pragma unroll 1
      for (int w = 0; w < 32; ++w) {
        const int c = waveCnt[w];
        if (w < wid) base += c;
        tot += c;
      }
      if (peak) outPos[base + pre] = (float)(cs + lt);
      running += tot;
      __syncthreads();  // protects waveCnt and cur for next round/chunk
    }
  }
  if (tid == 0) counts[sig * ROWS + row] = running;
}

// ---------------------------------------------------------------------------
// Kernel B: per-row right-aligned |diff| sum. grid = 32 rows, block = 256.
// ---------------------------------------------------------------------------
__global__ __launch_bounds__(256) void onsets_rowmae_kernel(
    const float* __restrict__ posP, const float* __restrict__ posG,
    const int* __restrict__ counts, float* __restrict__ rowsums) {
  const int row = blockIdx.x;
  const int nP = counts[row];
  const int nG = counts[ROWS + row];
  const float* P = posP + (size_t)row * MAXP;
  const float* G = posG + (size_t)row * MAXP;
  const int K = nP > nG ? nP : nG;

  float sum = 0.f;
  for (int k = threadIdx.x; k < K; k += 256) {
    const float p = (k < nP) ? P[nP - 1 - k] : 0.f;
    const float g = (k < nG) ? G[nG - 1 - k] : 0.f;
    sum += fabsf(p - g);
  }
  __shared__ float sd[256];
  sd[threadIdx.x] = sum;
  __syncthreads();
  for (int s = 128; s > 0; s >>= 1) {
    if (threadIdx.x < (unsigned)s) sd[threadIdx.x] += sd[threadIdx.x + s];
    __syncthreads();
  }
  if (threadIdx.x == 0) rowsums[row] = sd[0];
}

// ---------------------------------------------------------------------------
// Kernel C: reduce 32 row sums with V_WMMA_F32_16X16X4_F32.
// A[m][k] = rowsum[2m+k] for k<2 (lanes 0-15 hold K=0,1; lanes 16-31 hold
// K=2,3 = zeros), B = ones(4x16), C = 0 => D[m][n] = rowsum[2m]+rowsum[2m+1].
// Total = sum_m D[m][0] via 16 readlanes. One full wave32, EXEC all-ones.
// ---------------------------------------------------------------------------
typedef float v2f __attribute__((ext_vector_type(2)));
typedef float v8f __attribute__((ext_vector_type(8)));

__global__ __launch_bounds__(32) void onsets_finalize_kernel(
    const float* __restrict__ rowsums, float* __restrict__ out) {
  const int lane = threadIdx.x;
  __shared__ float sh[64];
  sh[lane]      = rowsums[lane];
  sh[32 + lane] = 0.f;
  __syncthreads();

  v2f a;
  a.x = sh[2 * lane];        // lanes >=16 index 32..63 -> zeros (K=2,3)
  a.y = sh[2 * lane + 1];
  v2f b;
  b.x = 1.f; b.y = 1.f;      // ones matrix: layout-independent
  v8f c = {};
  v8f d = __builtin_amdgcn_wmma_f32_16x16x4_f32(
      /*neg_a=*/false, a, /*neg_b=*/false, b,
      /*c_mod=*/(short)0, c, /*reuse_a=*/false, /*reuse_b=*/false);

  float tot = 0.f;
#pragma unroll
  for (int r = 0; r < 8; ++r)
    tot += __shfl(d[r], 0, 32) + __shfl(d[r], 16, 32);  // D[r][0] + D[8+r][0]

  if (lane == 0) out[0] = tot / 4194304.0f;  // mean over B*L = 32*131072
}

// ---------------------------------------------------------------------------
extern "C" void kernel_launch(void* const* d_in, const int* in_sizes, int n_in,
                              void* d_out, int out_size, void* d_ws, size_t ws_size,
                              hipStream_t stream) {
  const float* preds  = (const float*)d_in[0];
  const float* labels = (const float*)d_in[1];
  float* out = (float*)d_out;

  // Workspace layout (floats): posP[32*MAXP] | posG[32*MAXP] | counts(64 int) | rowsums[32]
  float* wsf     = (float*)d_ws;
  float* posP    = wsf;
  float* posG    = wsf + (size_t)ROWS * MAXP;
  int*   counts  = (int*)(wsf + (size_t)2 * ROWS * MAXP);
  float* rowsums = wsf + (size_t)2 * ROWS * MAXP + 64;

  dim3 gridA(ROWS, 2);
  onsets_peaks_kernel<<<gridA, TPB, 0, stream>>>(preds, labels, posP, counts);
  onsets_rowmae_kernel<<<ROWS, 256, 0, stream>>>(posP, posG, counts, rowsums);
  onsets_finalize_kernel<<<1, 32, 0, stream>>>(rowsums, out);
}